// LossEvaluator_88364657148330
// MI455X (gfx1250) — compile-verified
//
#include <hip/hip_runtime.h>
#include <hip/hip_bf16.h>
#include <stdint.h>

// ---------------------------------------------------------------------------
// Sequential 512-step trading-sim scan. One thread per (sample,batch) column,
// per-currency state in registers. Per-step inputs staged into LDS with the
// CDNA5 Tensor Data Mover (double-buffered, one step ahead); baseline mean
// over samples via wave32 shuffles + LDS; ys outputs via nontemporal stores.
// ---------------------------------------------------------------------------

#define SEQ     512
#define NCUR    8
#define NSAMP   64
#define NBATCH  256
#define BTILE   2                    // batch columns per block
#define BLOCK   (NSAMP * BTILE)      // 128 threads
#define NGRID   (NBATCH / BTILE)     // 128 blocks
#define NWAVE   (BLOCK / 32)         // 4 waves
#define LEV     10.0f

// LDS per-step tile layout (float indices inside one buffer)
#define L_SMP   0                    // 3*8*128 = 3072 floats
#define L_XLP   3072                 // 3072
#define L_FRAC  6144                 // 8*128 = 1024
#define L_ZLP   7168                 // 128
#define L_RATES 7296                 // 2*8*2 = 32
#define L_ACR   7328                 // 8*2 = 16
#define L_TOT   7344

#if defined(__HIP_DEVICE_COMPILE__) && __has_builtin(__builtin_amdgcn_tensor_load_to_lds) && __has_builtin(__builtin_amdgcn_s_wait_tensorcnt)
#define USE_TDM 1
#else
#define USE_TDM 0
#endif

typedef unsigned int v4u_t __attribute__((ext_vector_type(4)));
typedef int          v8i_t __attribute__((ext_vector_type(8)));
typedef int          v4i_t __attribute__((ext_vector_type(4)));

#if USE_TDM
// Build a D# for a (tile0 x tile1 x tile2) 4-byte-element tile with row stride
// stride0 and plane stride stride1 (element units), DMA'd to lds_off_bytes.
__device__ __forceinline__ void tdm_tile_load(unsigned lds_off_bytes, const float* gsrc,
                                              unsigned tile0, unsigned tile1, unsigned tile2,
                                              unsigned stride0, unsigned stride1)
{
  unsigned long long ga = (unsigned long long)(uintptr_t)gsrc;
  // Group 0: count=1 | lds_addr | global_addr[56:0] | type=2
  v4u_t g0 = { 1u,
               lds_off_bytes,
               (unsigned)ga,
               (((unsigned)(ga >> 32)) & 0x1ffffffu) | (2u << 30) };
  // Group 1: data_size=4B; tensor_dims = tile dims (tile starts at tensor
  // origin, no OOB); tile dims; dim0/dim1 strides (strides < 2^32 here).
  v8i_t g1 = { (int)(2u << 16),                                          // ws=0, data_size=2 (4B)
               (int)((tile0 & 0xffffu) << 16),                           // tensor_dim0 lo16
               (int)((tile0 >> 16) | ((tile1 & 0xffffu) << 16)),         // tensor_dim0 hi | tensor_dim1 lo
               (int)((tile1 >> 16) | ((tile0 & 0xffffu) << 16)),         // tensor_dim1 hi | tile_dim0
               (int)((tile1 & 0xffffu) | ((tile2 & 0xffffu) << 16)),     // tile_dim1 | tile_dim2 (0=unused)
               (int)stride0,                                             // tensor_dim0_stride[31:0]
               (int)((stride1 & 0xffffu) << 16),                         // s0 hi16 (=0) | tensor_dim1_stride lo16
               (int)(stride1 >> 16) };                                   // tensor_dim1_stride[47:16]
  v4i_t g2 = { (int)(tile2 ? tile2 : 1u), 0, 0, 0 };                     // tensor_dim2; no iterate/4D
  v4i_t g3 = { 0, 0, 0, 0 };
#if __clang_major__ >= 23
  v8i_t g4 = { 0, 0, 0, 0, 0, 0, 0, 0 };
  __builtin_amdgcn_tensor_load_to_lds(g0, g1, g2, g3, g4, 0);
#else
  __builtin_amdgcn_tensor_load_to_lds(g0, g1, g2, g3, 0);
#endif
}
#endif

// Issue (or, in the fallback, synchronously perform) the staging of step t's
// inputs for this block into LDS buffer `buf`.
__device__ __forceinline__ void stage_issue(float* buf, int t, int b0,
    const float* __restrict__ samples, const float* __restrict__ fractions,
    const float* __restrict__ xlp, const float* __restrict__ zlp,
    const float* __restrict__ rates, const float* __restrict__ acr)
{
#if USE_TDM
  if (threadIdx.x < 32) {  // wave 0 issues; TDM ignores EXEC, one issue per op
    unsigned off = (unsigned)(uintptr_t)buf;
    tdm_tile_load(off + L_SMP  * 4u, samples   + (size_t)t * 393216 + b0, BTILE, 64, 24, 256, 16384);
    tdm_tile_load(off + L_XLP  * 4u, xlp       + (size_t)t * 393216 + b0, BTILE, 64, 24, 256, 16384);
    tdm_tile_load(off + L_FRAC * 4u, fractions + (size_t)t * 131072 + b0, BTILE, 64, 8,  256, 16384);
    tdm_tile_load(off + L_ZLP  * 4u, zlp       + (size_t)t * 16384  + b0, BTILE, 64, 0,  256, 0);
    tdm_tile_load(off + L_RATES* 4u, rates     + (size_t)t * 4096   + b0, BTILE, 16, 0,  256, 0);
    tdm_tile_load(off + L_ACR  * 4u, acr       + (size_t)t * 2048   + b0, BTILE, 8,  0,  256, 0);
  }
#else
  const int tid = threadIdx.x;
  for (int i = tid; i < 3072; i += BLOCK) {
    int z = i >> 7, rem = i & 127;
    size_t src = (size_t)t * 393216 + (size_t)z * 16384 + (size_t)(rem >> 1) * 256 + b0 + (rem & 1);
    buf[L_SMP + i] = samples[src];
    buf[L_XLP + i] = xlp[src];
  }
  for (int i = tid; i < 1024; i += BLOCK) {
    int z = i >> 7, rem = i & 127;
    buf[L_FRAC + i] = fractions[(size_t)t * 131072 + (size_t)z * 16384 + (size_t)(rem >> 1) * 256 + b0 + (rem & 1)];
  }
  buf[L_ZLP + tid] = zlp[(size_t)t * 16384 + (size_t)(tid >> 1) * 256 + b0 + (tid & 1)];
  if (tid < 32) buf[L_RATES + tid] = rates[(size_t)t * 4096 + (size_t)(tid >> 1) * 256 + b0 + (tid & 1)];
  if (tid < 16) buf[L_ACR + tid]   = acr[(size_t)t * 2048 + (size_t)(tid >> 1) * 256 + b0 + (tid & 1)];
#endif
}

__device__ __forceinline__ void stage_wait(bool more_pending)
{
#if USE_TDM
  if (threadIdx.x < 32) {
    if (more_pending) __builtin_amdgcn_s_wait_tensorcnt(6);  // allow next step's 6 ops in flight
    else              __builtin_amdgcn_s_wait_tensorcnt(0);
  }
#else
  (void)more_pending;
#endif
}

__global__ __launch_bounds__(BLOCK, 1) void loss_scan_kernel(
    const float* __restrict__ samples, const float* __restrict__ fractions,
    const float* __restrict__ x_logprobs, const float* __restrict__ z_logprobs,
    const float* __restrict__ rates, const float* __restrict__ acr,
    const int* __restrict__ pos_states0, const int* __restrict__ pos_types0,
    const float* __restrict__ total_margin0, const float* __restrict__ open_sizes0,
    const float* __restrict__ open_rates0, float* __restrict__ out)
{
  __shared__ float lds[2][L_TOT];          // double-buffered staged inputs
  __shared__ float red_part[NWAVE * 16];   // per-wave partial sums (cur x b)
  __shared__ float red_base[16];           // baselines (cur x b)

  const int tid = threadIdx.x;
  const int bl  = tid & 1;                 // batch column within tile
  const int b0  = blockIdx.x * BTILE;
  const int col = (tid >> 1) * NBATCH + b0 + bl;   // n*256 + b

  // ---- carry state (registers) ----
  int   pstate[NCUR], ptype[NCUR];
  float open_sz[NCUR], open_rt[NCUR];
  float osize_lp[NCUR], cum_exec[NCUR], cum_close[NCUR], otype_lp[NCUR];
#pragma unroll
  for (int c = 0; c < NCUR; ++c) {
    const int s = c * (NSAMP * NBATCH) + col;
    pstate[c]   = pos_states0[s];
    ptype[c]    = pos_types0[s];
    open_sz[c]  = open_sizes0[s];
    open_rt[c]  = open_rates0[s];
    osize_lp[c] = 0.f; cum_exec[c] = 0.f; cum_close[c] = 0.f; otype_lp[c] = 0.f;
  }
  float tmargin = total_margin0[col];
  float cum_z = 0.f, loss = 0.f;

  const size_t O_PS  = 16384;
  const size_t O_PT  = O_PS + (size_t)SEQ * NCUR * NSAMP * NBATCH;
  const size_t O_TM  = O_PT + (size_t)SEQ * NCUR * NSAMP * NBATCH;
  const size_t O_OSZ = O_TM + (size_t)SEQ * NSAMP * NBATCH;
  const size_t O_ORT = O_OSZ + (size_t)SEQ * NCUR * NSAMP * NBATCH;

  stage_issue(&lds[0][0], 0, b0, samples, fractions, x_logprobs, z_logprobs, rates, acr);

  for (int t = 0; t < SEQ; ++t) {
    float* B = &lds[t & 1][0];
    const bool more = (t + 1) < SEQ;
    if (more)  // target buffer last read before previous step's final barrier
      stage_issue(&lds[(t + 1) & 1][0], t + 1, b0, samples, fractions, x_logprobs, z_logprobs, rates, acr);
    stage_wait(more);
    __syncthreads();                                   // [A] step-t data visible

    float rate0[NCUR], rate1[NCUR], acrv[NCUR];
#pragma unroll
    for (int c = 0; c < NCUR; ++c) {
      rate0[c] = B[L_RATES + c * 2 + bl];
      rate1[c] = B[L_RATES + 16 + c * 2 + bl];
      acrv[c]  = B[L_ACR + c * 2 + bl];
    }

    // ---- pre j-loop: P/L, closeout, masks, logprob bookkeeping ----
    float open_pl[NCUR], pl_sum = 0.f, acc_sum = 0.f;
#pragma unroll
    for (int c = 0; c < NCUR; ++c) {
      const bool  open_m = (pstate[c] == 1);
      const float cr = open_m ? ((ptype[c] == 0) ? rate0[c] : rate1[c]) : 1.0f;
      const float ao = open_sz[c] / acrv[c];
      open_pl[c] = LEV * ao * (1.0f - open_rt[c] / cr);
      pl_sum += open_pl[c];
      acc_sum += ao;
    }
    const bool closeout = (tmargin + pl_sum) < 0.5f * acc_sum;
    cum_z += B[L_ZLP + tid];

    bool  open_mask[NCUR], close_mask[NCUR];
    float mav_lp[NCUR], costs[NCUR];
    float S = 0.f;
#pragma unroll
    for (int c = 0; c < NCUR; ++c) S += cum_exec[c];
    float run = 0.f;
#pragma unroll
    for (int c = 0; c < NCUR; ++c) {
      const bool  open_m = (pstate[c] == 1);
      const float osmp = B[L_SMP + c * 128 + tid];          // samples[0] (open)
      const float csmp = B[L_SMP + 1024 + c * 128 + tid];   // samples[1] (close)
      open_mask[c]  = (!open_m) && (!closeout) && (osmp == 1.0f);
      close_mask[c] = open_m && (closeout || (csmp == 1.0f));
      if (open_mask[c] || close_mask[c]) pstate[c] ^= 1;
      const float olp = ((!closeout) && (!open_m)) ? B[L_XLP + c * 128 + tid] : 0.f;
      const float clp = ((!closeout) && open_m)    ? B[L_XLP + 1024 + c * 128 + tid] : 0.f;
      const float el  = olp + clp;
      run += el;
      mav_lp[c] = S + run;              // cum_exec.sum + inclusive cumsum
      cum_exec[c]  += el;
      cum_close[c] += clp;
      if (open_mask[c]) {
        otype_lp[c] = B[L_XLP + 2048 + c * 128 + tid];      // type_lp
        ptype[c]    = (int)B[L_SMP + 2048 + c * 128 + tid]; // type_s
        open_rt[c]  = (ptype[c] == 0) ? rate1[c] : rate0[c];
      }
      costs[c] = close_mask[c] ? open_pl[c] : 0.f;
    }

    // ---- baseline = mean over 64 samples, per (batch col, currency) ----
    float cs[NCUR];
#pragma unroll
    for (int c = 0; c < NCUR; ++c) cs[c] = costs[c];
#pragma unroll
    for (int m = 2; m < 32; m <<= 1) {
#pragma unroll
      for (int c = 0; c < NCUR; ++c) cs[c] += __shfl_xor(cs[c], m, 32);
    }
    const int lane = tid & 31, wid = tid >> 5;
    if (lane < 2) {
#pragma unroll
      for (int c = 0; c < NCUR; ++c) red_part[wid * 16 + c * 2 + lane] = cs[c];
    }
    __syncthreads();                                   // [B]
    if (tid < 16) {
      float s = 0.f;
#pragma unroll
      for (int w = 0; w < NWAVE; ++w) s += red_part[w * 16 + tid];
      red_base[tid] = s * (1.0f / (float)NSAMP);
    }
    __syncthreads();                                   // [C]
    float baseline[NCUR];
#pragma unroll
    for (int c = 0; c < NCUR; ++c) baseline[c] = red_base[c * 2 + bl];

    // ---- sequential per-currency loop (margin coupling) ----
#pragma unroll
    for (int j = 0; j < NCUR; ++j) {
      const float mavail = tmargin - acc_sum;          // total_margin - sum(open_sz/acr)
      if (open_mask[j]) {
        const float fj = B[L_FRAC + j * 128 + tid];
        const float new_size = fj * mavail * acrv[j];
        acc_sum += fj * mavail - open_sz[j] / acrv[j]; // new_size/acr[j] = fj*mavail
        open_sz[j]  = new_size;
        osize_lp[j] = mav_lp[j];
      }
      if (close_mask[j]) {
        const float cost_lp = cum_z + osize_lp[j] + otype_lp[j] + cum_close[j];
        loss += cost_lp * (costs[j] - baseline[j]) + costs[j];
        acc_sum -= open_sz[j] / acrv[j];
        open_sz[j] = 0.f;
        tmargin += open_pl[j];
      }
    }
#pragma unroll
    for (int c = 0; c < NCUR; ++c) if (close_mask[c]) cum_close[c] = 0.f;

    // ---- per-step ys outputs (post-update carry), nontemporal ----
    const size_t t4 = (size_t)t * (NCUR * NSAMP * NBATCH) + col;
#pragma unroll
    for (int c = 0; c < NCUR; ++c) {
      const size_t o = t4 + (size_t)c * (NSAMP * NBATCH);
      __builtin_nontemporal_store((float)pstate[c], out + O_PS + o);
      __builtin_nontemporal_store((float)ptype[c],  out + O_PT + o);
      __builtin_nontemporal_store(open_sz[c],       out + O_OSZ + o);
      __builtin_nontemporal_store(open_rt[c],       out + O_ORT + o);
    }
    __builtin_nontemporal_store(tmargin, out + O_TM + (size_t)t * (NSAMP * NBATCH) + col);

    __syncthreads();                                   // [D] fence before buffer reuse
  }

  out[col] = loss;                                     // loss is first output
}

extern "C" void kernel_launch(void* const* d_in, const int* in_sizes, int n_in,
                              void* d_out, int out_size, void* d_ws, size_t ws_size,
                              hipStream_t stream)
{
  (void)in_sizes; (void)n_in; (void)out_size; (void)d_ws; (void)ws_size;
  const float* samples   = (const float*)d_in[0];
  const float* fractions = (const float*)d_in[1];
  const float* x_lp      = (const float*)d_in[2];
  const float* z_lp      = (const float*)d_in[3];
  const float* rates     = (const float*)d_in[4];
  const float* acr       = (const float*)d_in[5];
  const int*   ps0       = (const int*)d_in[6];
  const int*   pt0       = (const int*)d_in[7];
  const float* tm0       = (const float*)d_in[8];
  const float* osz0      = (const float*)d_in[9];
  const float* ort0      = (const float*)d_in[10];
  float* out = (float*)d_out;

  loss_scan_kernel<<<dim3(NGRID), dim3(BLOCK), 0, stream>>>(
      samples, fractions, x_lp, z_lp, rates, acr, ps0, pt0, tm0, osz0, ort0, out);
}